// Diversity3_52699248721935
// MI455X (gfx1250) — compile-verified
//
#include <hip/hip_runtime.h>

typedef __attribute__((ext_vector_type(2))) float v2f;
typedef __attribute__((ext_vector_type(8))) float v8f;

// wave32 butterfly sum (all 32 lanes active)
__device__ __forceinline__ float waveSum(float v) {
#pragma unroll
    for (int m = 16; m >= 1; m >>= 1) v += __shfl_xor(v, m, 32);
    return v;
}

// One wave (32 lanes) per row. Streams the three model rows once,
// accumulates F_m = sum f_m and G_mn = sum f_m*f_n with f = exp(x/T)-1,
// then cos_mn = (C*G_mn - F_m*F_n)/sqrt((C*G_mm - F_m^2)(C*G_nn - F_n^2)).
__global__ __launch_bounds__(256) void div3_rows(const float* __restrict__ o1,
                                                 const float* __restrict__ o2,
                                                 const float* __restrict__ o3,
                                                 float* __restrict__ rowd,
                                                 int B, int C, float invT)
{
    const int wv   = threadIdx.x >> 5;
    const int lane = threadIdx.x & 31;
    const int row  = blockIdx.x * 8 + wv;
    if (row >= B) return;
    const long long base = (long long)row * (long long)C;

    float F1 = 0.f, F2 = 0.f, F3 = 0.f;
    float G11 = 0.f, G22 = 0.f, G33 = 0.f, G12 = 0.f, G13 = 0.f, G23 = 0.f;

    if ((C & 3) == 0) {
        const int C4 = C >> 2;
        const float4* __restrict__ p1 = (const float4*)(o1 + base);
        const float4* __restrict__ p2 = (const float4*)(o2 + base);
        const float4* __restrict__ p3 = (const float4*)(o3 + base);
        for (int i = lane; i < C4; i += 32) {
            float4 x1 = p1[i], x2 = p2[i], x3 = p3[i];
            float a1[4] = {x1.x, x1.y, x1.z, x1.w};
            float a2[4] = {x2.x, x2.y, x2.z, x2.w};
            float a3[4] = {x3.x, x3.y, x3.z, x3.w};
#pragma unroll
            for (int k = 0; k < 4; ++k) {
                float f1 = __expf(a1[k] * invT) - 1.0f;
                float f2 = __expf(a2[k] * invT) - 1.0f;
                float f3 = __expf(a3[k] * invT) - 1.0f;
                F1 += f1; F2 += f2; F3 += f3;
                G11 = fmaf(f1, f1, G11);
                G22 = fmaf(f2, f2, G22);
                G33 = fmaf(f3, f3, G33);
                G12 = fmaf(f1, f2, G12);
                G13 = fmaf(f1, f3, G13);
                G23 = fmaf(f2, f3, G23);
            }
        }
    } else {
        for (int i = lane; i < C; i += 32) {
            float f1 = __expf(o1[base + i] * invT) - 1.0f;
            float f2 = __expf(o2[base + i] * invT) - 1.0f;
            float f3 = __expf(o3[base + i] * invT) - 1.0f;
            F1 += f1; F2 += f2; F3 += f3;
            G11 = fmaf(f1, f1, G11);
            G22 = fmaf(f2, f2, G22);
            G33 = fmaf(f3, f3, G33);
            G12 = fmaf(f1, f2, G12);
            G13 = fmaf(f1, f3, G13);
            G23 = fmaf(f2, f3, G23);
        }
    }

    F1 = waveSum(F1);  F2 = waveSum(F2);  F3 = waveSum(F3);
    G11 = waveSum(G11); G22 = waveSum(G22); G33 = waveSum(G33);
    G12 = waveSum(G12); G13 = waveSum(G13); G23 = waveSum(G23);

    if (lane == 0) {
        const float Cf = (float)C;
        float n1  = fmaf(Cf, G11, -F1 * F1);
        float n2  = fmaf(Cf, G22, -F2 * F2);
        float n3  = fmaf(Cf, G33, -F3 * F3);
        float d12 = fmaf(Cf, G12, -F1 * F2);
        float d13 = fmaf(Cf, G13, -F1 * F3);
        float d23 = fmaf(Cf, G23, -F2 * F3);
        float c12 = d12 * rsqrtf(n1 * n2);
        float c13 = d13 * rsqrtf(n1 * n3);
        float c23 = d23 * rsqrtf(n2 * n3);
        rowd[row] = c12 + c13 + c23;
    }
}

// Deterministic final reduction. 256 threads accumulate strided partials,
// then wave 0 sums the 256 partials with V_WMMA_F32_16X16X4_F32:
// A = all-ones 16x4, so D = A*B + C adds every element of B exactly once to
// each of D's 16 identical rows -> grand total = (sum of all D elements)/16.
// Exact f32 adds, fixed order, layout-independent.
__global__ __launch_bounds__(256) void div3_reduce(const float* __restrict__ rowd,
                                                   float* __restrict__ out,
                                                   int n, float scaleOverN)
{
    __shared__ float part[256];
    const int t = threadIdx.x;
    float s = 0.f;
    for (int i = t; i < n; i += 256) s += rowd[i];
    part[t] = s;
    __syncthreads();
    if (t >= 32) return;   // wave 0 continues with EXEC all-ones (WMMA requirement)

    v2f a; a.x = 1.0f; a.y = 1.0f;     // 16x4 all-ones A matrix
    v8f acc = {};
#pragma unroll
    for (int q = 0; q < 4; ++q) {
        v2f b;
        b.x = part[q * 64 + 2 * t];
        b.y = part[q * 64 + 2 * t + 1];
        acc = __builtin_amdgcn_wmma_f32_16x16x4_f32(
            /*neg_a=*/false, a, /*neg_b=*/false, b,
            /*c_mod=*/(short)0, acc, /*reuse_a=*/false, /*reuse_b=*/false);
    }
    float v = acc[0] + acc[1] + acc[2] + acc[3] + acc[4] + acc[5] + acc[6] + acc[7];
#pragma unroll
    for (int m = 16; m >= 1; m >>= 1) v += __shfl_xor(v, m, 32);
    if (t == 0) out[0] = scaleOverN * (v * 0.0625f);   // /16: D has 16 identical rows
}

extern "C" void kernel_launch(void* const* d_in, const int* in_sizes, int n_in,
                              void* d_out, int out_size, void* d_ws, size_t ws_size,
                              hipStream_t stream) {
    const float* o1 = (const float*)d_in[0];
    const float* o2 = (const float*)d_in[1];
    const float* o3 = (const float*)d_in[2];
    // d_in[3] (targets) is unused by the reference math.

    const int B = in_sizes[3];              // 16384
    const int C = in_sizes[0] / B;          // 1000
    const float invT = 1.0f / 20.0f;
    const float SCALE = 0.3f;

    float* rowd = (float*)d_ws;             // B floats of scratch
    float* out  = (float*)d_out;

    const int blocks = (B + 7) / 8;         // 8 waves (rows) per 256-thread block
    div3_rows<<<blocks, 256, 0, stream>>>(o1, o2, o3, rowd, B, C, invT);
    div3_reduce<<<1, 256, 0, stream>>>(rowd, out, B, SCALE / (float)B);
}